// TGCN_84086869721572
// MI455X (gfx1250) — compile-verified
//
#include <hip/hip_runtime.h>
#include <math.h>

typedef float v2f __attribute__((ext_vector_type(2)));
typedef float v8f __attribute__((ext_vector_type(8)));

#define T_STEPS 128
#define NN      24576
#define IN_F    32
#define GH      16
#define NG      512
#define NE      8192
#define HID     256
#define GATE3   768

// sigmoid via v_exp_f32 + v_rcp_f32 (no IEEE divide chain)
__device__ __forceinline__ float sigmoidf_(float x) {
    float e = __expf(-x);
    return __builtin_amdgcn_rcpf(1.0f + e);
}
// tanh(x) = 1 - 2/(1 + e^{2x}); saturates correctly at +/-inf
__device__ __forceinline__ float tanhf_(float x) {
    float e = __expf(2.0f * x);
    return 1.0f - 2.0f * __builtin_amdgcn_rcpf(1.0f + e);
}

// ---------------- K1: degree + per-edge symmetric norms --------------------
__global__ void prep_kernel(const int* __restrict__ ei, float* __restrict__ dinv,
                            float* __restrict__ norm) {
    __shared__ int   cnt[NG];
    __shared__ float sdv[NG];
    int tid = threadIdx.x;                 // 512 threads, 1 block
    cnt[tid] = 0;
    __syncthreads();
    for (int e = tid; e < NE; e += NG) atomicAdd(&cnt[ei[NE + e]], 1);
    __syncthreads();
    float dv = rsqrtf(1.0f + (float)cnt[tid]);   // +1 self loop
    sdv[tid] = dv;
    dinv[tid] = dv;
    __syncthreads();
    for (int e = tid; e < NE; e += NG)
        norm[e] = sdv[ei[e]] * sdv[ei[NE + e]];
}

// ---------------- K2: exact 2-layer GCN for nodes < 512 (one block per t) --
__global__ void gcn_small_kernel(const float* __restrict__ x, const int* __restrict__ ei,
                                 const float* __restrict__ w1, const float* __restrict__ b1,
                                 const float* __restrict__ w2, const float* __restrict__ b2,
                                 const float* __restrict__ dinv, const float* __restrict__ norm,
                                 float* __restrict__ g_small, float* __restrict__ g0) {
    __shared__ float sA[NG * GH];   // h1, then z1 (in place)
    __shared__ float sB[NG * GH];   // agg1; later [0..1535]=h2, [4096..5631]=agg2
    int tid = threadIdx.x;          // 256 threads
    int t   = blockIdx.x;

    for (int i = tid; i < NG * GH; i += 256) sB[i] = 0.0f;
    __syncthreads();

    // layer-1 matmul for the 512 graph nodes
    const float* xt = x + (size_t)t * NN * IN_F;
    for (int n = tid; n < NG; n += 256) {
        float xr[IN_F];
        const float* row = xt + n * IN_F;
        #pragma unroll
        for (int k = 0; k < IN_F; k++) xr[k] = row[k];
        for (int f = 0; f < GH; f++) {
            float s = 0.0f;
            #pragma unroll
            for (int k = 0; k < IN_F; k++) s += xr[k] * w1[k * GH + f];
            sA[n * GH + f] = s;
        }
    }
    __syncthreads();

    // edge aggregation, layer 1
    for (int e = tid; e < NE; e += 256) {
        int s = ei[e], d = ei[NE + e];
        float w = norm[e];
        #pragma unroll
        for (int f = 0; f < GH; f++)
            atomicAdd(&sB[d * GH + f], sA[s * GH + f] * w);
    }
    __syncthreads();

    // z1 = sigmoid(agg + self + b1)  (in place over sA)
    for (int i = tid; i < NG * GH; i += 256) {
        int n = i >> 4, f = i & 15;
        float dv = dinv[n];
        sA[i] = sigmoidf_(sB[i] + sA[i] * dv * dv + b1[f]);
    }
    __syncthreads();

    // h2 = z1 @ W2 into sB[0..1535]; zero agg2 region sB[4096..5631]
    for (int i = tid; i < NG * 3; i += 256) {
        int n = i / 3, c = i % 3;
        float s = 0.0f;
        #pragma unroll
        for (int f = 0; f < GH; f++) s += sA[n * GH + f] * w2[f * 3 + c];
        sB[i] = s;
        sB[4096 + i] = 0.0f;
    }
    __syncthreads();

    // edge aggregation, layer 2
    for (int e = tid; e < NE; e += 256) {
        int s = ei[e], d = ei[NE + e];
        float w = norm[e];
        #pragma unroll
        for (int c = 0; c < 3; c++)
            atomicAdd(&sB[4096 + d * 3 + c], sB[s * 3 + c] * w);
    }
    __syncthreads();

    for (int i = tid; i < NG * 3; i += 256) {
        int n = i / 3, c = i % 3;
        float dv = dinv[n];
        float g = sigmoidf_(sB[4096 + i] + sB[i] * dv * dv + b2[c]);
        g_small[(size_t)t * NG * 3 + i] = g;
        if (c == 0) g0[(size_t)t * NN + n] = g;
    }
}

// ---------------- K3: elementwise path for nodes >= 512 (WMMA f32) ---------
#define XS_STRIDE 36
#define ZT_STRIDE 17

__global__ void gcn_big_kernel(const float* __restrict__ x,
                               const float* __restrict__ w1, const float* __restrict__ b1,
                               const float* __restrict__ w2, const float* __restrict__ b2,
                               float* __restrict__ g0, float* __restrict__ out_mu,
                               float* __restrict__ out_lv) {
    __shared__ float xs[4][16 * XS_STRIDE];
    __shared__ float zt[4][16 * ZT_STRIDE];
    int lane = threadIdx.x & 31;
    int wv   = threadIdx.x >> 5;
    int gw   = blockIdx.x * 4 + wv;              // 0..1503
    int n0   = NG + gw * 16;
    int r    = lane & 15, hi = lane >> 4;

    // B fragments of W1 (K=32 -> 8 k-steps), resident in VGPRs
    v2f bf[8];
    #pragma unroll
    for (int kk = 0; kk < 8; kk++) {
        int k = kk * 4 + 2 * hi;
        bf[kk].x = w1[k * GH + r];
        bf[kk].y = w1[(k + 1) * GH + r];
    }
    float bias1 = b1[r];
    float w2r[GH * 3];
    #pragma unroll
    for (int i = 0; i < GH * 3; i++) w2r[i] = w2[i];
    float b20 = b2[0], b21 = b2[1], b22 = b2[2];
    float s1 = 0.0f, s2 = 0.0f;
    float* xsw = xs[wv];
    float* ztw = zt[wv];

    for (int t = 0; t < T_STEPS; t++) {
        const float* base = x + ((size_t)t * NN + n0) * IN_F;   // 512 contiguous floats
        #pragma unroll
        for (int i = 0; i < 4; i++) {
            int flat = lane * 4 + i * 128;
            int row = flat >> 5, col = flat & 31;
            float4 v = *(const float4*)(base + flat);
            *(float4*)&xsw[row * XS_STRIDE + col] = v;
        }
        // prefetch next timestep's tile (global_prefetch_b8): one address per
        // 64B line per lane covers the whole 2KB tile
        if (t + 1 < T_STEPS)
            __builtin_prefetch(base + (size_t)NN * IN_F + lane * 16, 0, 1);
        __syncthreads();
        v8f acc = {0.f, 0.f, 0.f, 0.f, 0.f, 0.f, 0.f, 0.f};
        #pragma unroll
        for (int kk = 0; kk < 8; kk++) {
            int k = kk * 4 + 2 * hi;
            v2f a = *(const v2f*)&xsw[r * XS_STRIDE + k];
            acc = __builtin_amdgcn_wmma_f32_16x16x4_f32(false, a, false, bf[kk],
                                                        (short)0, acc, false, false);
        }
        #pragma unroll
        for (int v = 0; v < 8; v++)
            ztw[(v + 8 * hi) * ZT_STRIDE + r] = sigmoidf_(acc[v] + bias1);
        __syncthreads();
        if (lane < 16) {
            float g0v = b20, g1v = b21, g2v = b22;
            #pragma unroll
            for (int f = 0; f < GH; f++) {
                float z = ztw[lane * ZT_STRIDE + f];
                g0v += z * w2r[f * 3 + 0];
                g1v += z * w2r[f * 3 + 1];
                g2v += z * w2r[f * 3 + 2];
            }
            g0[(size_t)t * NN + n0 + lane] = sigmoidf_(g0v);
            s1 += sigmoidf_(g1v);
            s2 += sigmoidf_(g2v);
        }
        __syncthreads();
    }
    if (lane < 16) {
        out_mu[n0 + lane] = s1 * (1.0f / T_STEPS);
        out_lv[n0 + lane] = s2 * (1.0f / T_STEPS);
    }
}

// ---------------- K4: C(M,N) = A(M,K) @ B(N,K)^T + bias(N), WMMA f32 -------
__global__ void gemm_xt_bias_kernel(const float* __restrict__ A, const float* __restrict__ B,
                                    const float* __restrict__ bias, float* __restrict__ C,
                                    int M, int N, int K) {
    int lane = threadIdx.x & 31;
    int wv   = threadIdx.x >> 5;
    int gw   = blockIdx.x * 2 + wv;
    int ntiles = N >> 4;
    int mt = gw / ntiles, nt = gw % ntiles;
    int r = lane & 15, hi = lane >> 4;
    const float* arow = A + (size_t)(mt * 16 + r) * K + 2 * hi;
    const float* brow = B + (size_t)(nt * 16 + r) * K + 2 * hi;
    v8f acc = {0.f, 0.f, 0.f, 0.f, 0.f, 0.f, 0.f, 0.f};
    for (int k0 = 0; k0 < K; k0 += 4) {
        v2f a = *(const v2f*)(arow + k0);
        v2f b = *(const v2f*)(brow + k0);
        acc = __builtin_amdgcn_wmma_f32_16x16x4_f32(false, a, false, b,
                                                    (short)0, acc, false, false);
    }
    float bs = bias[nt * 16 + r];
    #pragma unroll
    for (int v = 0; v < 8; v++)
        C[(size_t)(mt * 16 + v + 8 * hi) * N + nt * 16 + r] = acc[v] + bs;
}

// ---------------- K5: GRU recurrent scan (one layer), single workgroup -----
#define H_STRIDE 260
__global__ void gru_scan_kernel(const float* __restrict__ xp, const float* __restrict__ whh,
                                const float* __restrict__ bhh, float* __restrict__ hseq) {
    __shared__ float h_lds[16 * H_STRIDE];
    int tid  = threadIdx.x;           // 512 threads = 16 waves
    int lane = tid & 31, wv = tid >> 5;
    int r    = lane & 15, hi = lane >> 4;
    int j    = wv * 16 + r;           // hidden column 0..255
    for (int i = tid; i < 16 * H_STRIDE; i += 512) h_lds[i] = 0.0f;
    float bj_r = bhh[j], bj_z = bhh[HID + j], bj_n = bhh[2 * HID + j];
    const float* wr = whh + (size_t)j * HID + 2 * hi;
    const float* wz = whh + (size_t)(HID + j) * HID + 2 * hi;
    const float* wn = whh + (size_t)(2 * HID + j) * HID + 2 * hi;
    __syncthreads();

    for (int t = 0; t < T_STEPS; t++) {
        v8f ar = {0.f,0.f,0.f,0.f,0.f,0.f,0.f,0.f};
        v8f az = {0.f,0.f,0.f,0.f,0.f,0.f,0.f,0.f};
        v8f an = {0.f,0.f,0.f,0.f,0.f,0.f,0.f,0.f};
        for (int k0 = 0; k0 < HID; k0 += 4) {
            v2f a  = *(const v2f*)&h_lds[r * H_STRIDE + k0 + 2 * hi];
            v2f b0 = *(const v2f*)(wr + k0);
            v2f b1 = *(const v2f*)(wz + k0);
            v2f b2 = *(const v2f*)(wn + k0);
            ar = __builtin_amdgcn_wmma_f32_16x16x4_f32(false, a, false, b0, (short)0, ar, false, false);
            az = __builtin_amdgcn_wmma_f32_16x16x4_f32(false, a, false, b1, (short)0, az, false, false);
            an = __builtin_amdgcn_wmma_f32_16x16x4_f32(false, a, false, b2, (short)0, an, false, false);
        }
        __syncthreads();   // everyone done reading old h
        const float* xpt = xp + (size_t)t * 16 * GATE3;
        #pragma unroll
        for (int v = 0; v < 8; v++) {
            int b = v + 8 * hi;       // batch row
            float xr = xpt[b * GATE3 + j];
            float xz = xpt[b * GATE3 + HID + j];
            float xn = xpt[b * GATE3 + 2 * HID + j];
            float rg = sigmoidf_(xr + ar[v] + bj_r);
            float zg = sigmoidf_(xz + az[v] + bj_z);
            float ng = tanhf_(xn + rg * (an[v] + bj_n));
            float hold = h_lds[b * H_STRIDE + j];
            float hn = (1.0f - zg) * ng + zg * hold;
            h_lds[b * H_STRIDE + j] = hn;
            hseq[((size_t)t * 16 + b) * HID + j] = hn;
        }
        __syncthreads();   // h updated before next step's WMMA reads
    }
}

// ---------------- K6: final linear ----------------------------------------
__global__ void final_linear_kernel(const float* __restrict__ h2seq,
                                    const float* __restrict__ lw, const float* __restrict__ lb,
                                    float* __restrict__ out) {
    int i = blockIdx.x * 256 + threadIdx.x;     // 0..2047
    if (i >= T_STEPS * 16) return;
    const float* hrow = h2seq + (size_t)i * HID;
    float s = lb[0];
    for (int k = 0; k < HID; k++) s += hrow[k] * lw[k];
    out[i] = s;
}

// ---------------- K7: mu/logvar for nodes < 512 ----------------------------
__global__ void small_stats_kernel(const float* __restrict__ g_small,
                                   float* __restrict__ mu, float* __restrict__ lv) {
    int n = blockIdx.x * 256 + threadIdx.x;
    if (n >= NG) return;
    float s1 = 0.0f, s2 = 0.0f;
    for (int t = 0; t < T_STEPS; t++) {
        s1 += g_small[(size_t)t * NG * 3 + n * 3 + 1];
        s2 += g_small[(size_t)t * NG * 3 + n * 3 + 2];
    }
    mu[n] = s1 * (1.0f / T_STEPS);
    lv[n] = s2 * (1.0f / T_STEPS);
}

extern "C" void kernel_launch(void* const* d_in, const int* in_sizes, int n_in,
                              void* d_out, int out_size, void* d_ws, size_t ws_size,
                              hipStream_t stream) {
    const float* x    = (const float*)d_in[0];
    const int*   ei   = (const int*)d_in[1];
    const float* w1   = (const float*)d_in[2];
    const float* b1   = (const float*)d_in[3];
    const float* w2   = (const float*)d_in[4];
    const float* b2   = (const float*)d_in[5];
    const float* wih0 = (const float*)d_in[6];
    const float* whh0 = (const float*)d_in[7];
    const float* bih0 = (const float*)d_in[8];
    const float* bhh0 = (const float*)d_in[9];
    const float* wih1 = (const float*)d_in[10];
    const float* whh1 = (const float*)d_in[11];
    const float* bih1 = (const float*)d_in[12];
    const float* bhh1 = (const float*)d_in[13];
    const float* lw   = (const float*)d_in[14];
    const float* lb   = (const float*)d_in[15];

    float* out     = (float*)d_out;        // [2048] out, [24576] mu, [24576] logvar
    float* out_mu  = out + 2048;
    float* out_lv  = out + 2048 + NN;

    float* ws      = (float*)d_ws;
    float* g0      = ws;  ws += (size_t)T_STEPS * NN;        // 3,145,728
    float* g_small = ws;  ws += (size_t)T_STEPS * NG * 3;    //   196,608
    float* dinv    = ws;  ws += NG;                          //       512
    float* norm    = ws;  ws += NE;                          //     8,192
    float* xp0     = ws;  ws += (size_t)T_STEPS * 16 * GATE3;// 1,572,864
    float* h1seq   = ws;  ws += (size_t)T_STEPS * 16 * HID;  //   524,288
    float* xp1     = ws;  ws += (size_t)T_STEPS * 16 * GATE3;// 1,572,864
    float* h2seq   = ws;  /* 524,288 */                      // ~28.8 MB total

    prep_kernel<<<1, 512, 0, stream>>>(ei, dinv, norm);
    gcn_small_kernel<<<T_STEPS, 256, 0, stream>>>(x, ei, w1, b1, w2, b2, dinv, norm,
                                                  g_small, g0);
    gcn_big_kernel<<<376, 128, 0, stream>>>(x, w1, b1, w2, b2, g0, out_mu, out_lv);
    small_stats_kernel<<<2, 256, 0, stream>>>(g_small, out_mu, out_lv);

    // xp0 = rnn_in @ w_ih_l0^T + b_ih_l0   (M=2048, N=768, K=1536)
    gemm_xt_bias_kernel<<<(2048 / 16) * (768 / 16) / 2, 64, 0, stream>>>(
        g0, wih0, bih0, xp0, 2048, 768, 1536);
    gru_scan_kernel<<<1, 512, 0, stream>>>(xp0, whh0, bhh0, h1seq);

    // xp1 = h1seq @ w_ih_l1^T + b_ih_l1    (M=2048, N=768, K=256)
    gemm_xt_bias_kernel<<<(2048 / 16) * (768 / 16) / 2, 64, 0, stream>>>(
        h1seq, wih1, bih1, xp1, 2048, 768, 256);
    gru_scan_kernel<<<1, 512, 0, stream>>>(xp1, whh1, bhh1, h2seq);

    final_linear_kernel<<<8, 256, 0, stream>>>(h2seq, lw, lb, out);
}